// BackboneSolver_23691039604765
// MI455X (gfx1250) — compile-verified
//
#include <hip/hip_runtime.h>
#include <hip/hip_bf16.h>
#include <stdint.h>

#define NREF 4096   // N nodes per batch (reference setup)
#define KP   64     // K neighbors per node

// ---------------- CDNA5 async global->LDS plumbing (compile-guarded) --------
#if defined(__gfx1250__) && __has_builtin(__builtin_amdgcn_global_load_async_to_lds_b128)
#define USE_ASYNC_LDS 1
#else
#define USE_ASYNC_LDS 0
#endif

#if USE_ASYNC_LDS
typedef int v4i_t __attribute__((ext_vector_type(4)));
typedef __attribute__((address_space(1))) v4i_t as1_v4i;  // global
typedef __attribute__((address_space(3))) v4i_t as3_v4i;  // LDS

__device__ __forceinline__ void async_wait_all() {
#if __has_builtin(__builtin_amdgcn_s_wait_asynccnt)
  __builtin_amdgcn_s_wait_asynccnt(0);
#else
  asm volatile("s_wait_asynccnt 0" ::: "memory");
#endif
}
#endif

// Copy NCHUNK 16-byte chunks from a 16B-aligned global row into LDS.
template <int NCHUNK>
__device__ __forceinline__ void stage_tile(const float* __restrict__ g, float* l, int tid) {
#pragma unroll
  for (int base = 0; base < NCHUNK; base += 64) {
    const int c = base + tid;
    if (c < NCHUNK) {
#if USE_ASYNC_LDS
      __builtin_amdgcn_global_load_async_to_lds_b128(
          (as1_v4i*)(g + 4 * c), (as3_v4i*)(l + 4 * c), 0, 0);
#else
      const float4 v = ((const float4*)g)[c];
      ((float4*)l)[c] = v;
#endif
    }
  }
}

// ---------------- small 3x3 helpers (row-major float[9]) --------------------
__device__ __forceinline__ void ld3(const float* p, float* v) {
  v[0] = p[0]; v[1] = p[1]; v[2] = p[2];
}
__device__ __forceinline__ void ld9(const float* p, float* v) {
#pragma unroll
  for (int i = 0; i < 9; ++i) v[i] = p[i];
}
__device__ __forceinline__ void m_mul(const float* A, const float* B, float* C) { // C = A@B
#pragma unroll
  for (int i = 0; i < 3; ++i)
#pragma unroll
    for (int j = 0; j < 3; ++j)
      C[i * 3 + j] = A[i * 3 + 0] * B[0 + j] + A[i * 3 + 1] * B[3 + j] + A[i * 3 + 2] * B[6 + j];
}
__device__ __forceinline__ void m_ATmul(const float* A, const float* B, float* C) { // C = A^T@B
#pragma unroll
  for (int i = 0; i < 3; ++i)
#pragma unroll
    for (int j = 0; j < 3; ++j)
      C[i * 3 + j] = A[0 + i] * B[0 + j] + A[3 + i] * B[3 + j] + A[6 + i] * B[6 + j];
}
__device__ __forceinline__ void m_mulBT(const float* A, const float* B, float* C) { // C = A@B^T
#pragma unroll
  for (int i = 0; i < 3; ++i)
#pragma unroll
    for (int j = 0; j < 3; ++j)
      C[i * 3 + j] = A[i * 3 + 0] * B[j * 3 + 0] + A[i * 3 + 1] * B[j * 3 + 1] + A[i * 3 + 2] * B[j * 3 + 2];
}
__device__ __forceinline__ void m_vec(const float* A, const float* v, float* o) { // o = A v
#pragma unroll
  for (int i = 0; i < 3; ++i) o[i] = A[i * 3 + 0] * v[0] + A[i * 3 + 1] * v[1] + A[i * 3 + 2] * v[2];
}
__device__ __forceinline__ void m_Tvec(const float* A, const float* v, float* o) { // o = A^T v
#pragma unroll
  for (int i = 0; i < 3; ++i) o[i] = A[0 + i] * v[0] + A[3 + i] * v[1] + A[6 + i] * v[2];
}
__device__ __forceinline__ float dot3(const float* a, const float* b) {
  return a[0] * b[0] + a[1] * b[1] + a[2] * b[2];
}
__device__ __forceinline__ void cross3(const float* a, const float* b, float* c) {
  c[0] = a[1] * b[2] - a[2] * b[1];
  c[1] = a[2] * b[0] - a[0] * b[2];
  c[2] = a[0] * b[1] - a[1] * b[0];
}

// Branchless cyclic Jacobi rotation on symmetric A (zeroing A[P][Q]); V = V*G.
template <int P, int Q>
__device__ __forceinline__ void jrot(float* A, float* V) {
  const float apq = A[P * 3 + Q];
  const float app = A[P * 3 + P], aqq = A[Q * 3 + Q];
  const bool ok = fabsf(apq) > 1e-30f;
  const float denom = ok ? (2.0f * apq) : 1.0f;
  const float tau = (aqq - app) / denom;
  const float tt = (tau >= 0.0f ? 1.0f : -1.0f) / (fabsf(tau) + sqrtf(1.0f + tau * tau));
  float c = 1.0f / sqrtf(1.0f + tt * tt);
  float s = tt * c;
  s = ok ? s : 0.0f;
  c = ok ? c : 1.0f;
#pragma unroll
  for (int k = 0; k < 3; ++k) { // columns
    const float akp = A[k * 3 + P], akq = A[k * 3 + Q];
    A[k * 3 + P] = c * akp - s * akq;
    A[k * 3 + Q] = s * akp + c * akq;
  }
#pragma unroll
  for (int k = 0; k < 3; ++k) { // rows
    const float apk = A[P * 3 + k], aqk = A[Q * 3 + k];
    A[P * 3 + k] = c * apk - s * aqk;
    A[Q * 3 + k] = s * apk + c * aqk;
  }
#pragma unroll
  for (int k = 0; k < 3; ++k) {
    const float vkp = V[k * 3 + P], vkq = V[k * 3 + Q];
    V[k * 3 + P] = c * vkp - s * vkq;
    V[k * 3 + Q] = s * vkp + c * vkq;
  }
}

__device__ __forceinline__ void getcol(const float* V, int i, float* v) {
#pragma unroll
  for (int r = 0; r < 3; ++r)
    v[r] = (i == 0) ? V[r * 3 + 0] : ((i == 1) ? V[r * 3 + 1] : V[r * 3 + 2]);
}

// Reference proj_svd: with M = U S V^T (svd), returns V diag(1,1,d) U^T,
// d = sign(det V * det U). Implemented via Jacobi on M^T M; the sign ambiguity
// in U's 3rd column cancels by using row2 = det(V) * (u0 x u1).
__device__ void proj_so3(const float* M, float* R) {
  float A[9];
  m_ATmul(M, M, A); // M^T M
  float V[9] = {1.f, 0.f, 0.f, 0.f, 1.f, 0.f, 0.f, 0.f, 1.f};
#pragma unroll
  for (int sw = 0; sw < 5; ++sw) {
    jrot<0, 1>(A, V);
    jrot<0, 2>(A, V);
    jrot<1, 2>(A, V);
  }
  const float l0 = A[0], l1 = A[4], l2 = A[8];
  int i0 = (l0 >= l1) ? ((l0 >= l2) ? 0 : 2) : ((l1 >= l2) ? 1 : 2);
  int i2 = (l0 <= l1) ? ((l0 <= l2) ? 0 : 2) : ((l1 <= l2) ? 1 : 2);
  if (i0 == i2) { i0 = 0; i2 = 2; }
  const int i1 = 3 - i0 - i2;
  float v0[3], v1[3], v2[3];
  getcol(V, i0, v0); getcol(V, i1, v1); getcol(V, i2, v2);

  float u0[3];
  m_vec(M, v0, u0);
  const float n0 = sqrtf(dot3(u0, u0));
  if (n0 < 1e-20f) { // fully degenerate input
#pragma unroll
    for (int i = 0; i < 9; ++i) R[i] = (i % 4 == 0) ? 1.f : 0.f;
    return;
  }
  const float r0 = 1.0f / n0;
#pragma unroll
  for (int i = 0; i < 3; ++i) u0[i] *= r0;

  float u1[3];
  m_vec(M, v1, u1);
  const float d01 = dot3(u1, u0);
#pragma unroll
  for (int i = 0; i < 3; ++i) u1[i] -= d01 * u0[i];
  const float n1 = sqrtf(dot3(u1, u1));
  if (n1 > 1e-20f) {
    const float r1 = 1.0f / n1;
#pragma unroll
    for (int i = 0; i < 3; ++i) u1[i] *= r1;
  } else { // rank-1: pick any unit vector orthogonal to u0
    float e[3] = {0.f, 0.f, 0.f};
    const float ax = fabsf(u0[0]), ay = fabsf(u0[1]), az = fabsf(u0[2]);
    if (ax <= ay && ax <= az) e[0] = 1.f;
    else if (ay <= az) e[1] = 1.f;
    else e[2] = 1.f;
    cross3(u0, e, u1);
    const float rn = 1.0f / sqrtf(dot3(u1, u1));
#pragma unroll
    for (int i = 0; i < 3; ++i) u1[i] *= rn;
  }
  float c2[3];
  cross3(u0, u1, c2);
  float cv[3];
  cross3(v1, v2, cv);
  const float dV = (dot3(v0, cv) >= 0.f) ? 1.f : -1.f;
#pragma unroll
  for (int i = 0; i < 3; ++i)
#pragma unroll
    for (int k = 0; k < 3; ++k)
      R[i * 3 + k] = v0[i] * u0[k] + v1[i] * u1[k] + dV * v2[i] * c2[k];
}

// Wave32 shuffle reduction + 2-wave LDS combine; thread 0 does SVD + store.
__device__ __forceinline__ void reduce_finish(float* red, int tid, float wk, const float* tv,
                                              const float* Om, float* t_dst, float* O_dst) {
  float vals[13];
  vals[0] = wk;
#pragma unroll
  for (int c = 0; c < 3; ++c) vals[1 + c] = wk * tv[c];
#pragma unroll
  for (int c = 0; c < 9; ++c) vals[4 + c] = wk * Om[c];
#pragma unroll
  for (int c = 0; c < 13; ++c) {
#pragma unroll
    for (int off = 16; off > 0; off >>= 1) vals[c] += __shfl_down(vals[c], off, 32);
  }
  const int lane = tid & 31;
  const int wave = tid >> 5;
  if (lane == 0) {
#pragma unroll
    for (int c = 0; c < 13; ++c) red[wave * 13 + c] = vals[c];
  }
  __syncthreads();
  if (tid == 0) {
    float tot[13];
#pragma unroll
    for (int c = 0; c < 13; ++c) tot[c] = red[c] + red[13 + c];
    const float inv = 1.0f / tot[0];
    float Ms[9];
#pragma unroll
    for (int c = 0; c < 9; ++c) Ms[c] = tot[4 + c] * inv;
    float R[9];
    proj_so3(Ms, R);
#pragma unroll
    for (int c = 0; c < 3; ++c) t_dst[c] = tot[1 + c] * inv;
#pragma unroll
    for (int c = 0; c < 9; ++c) O_dst[c] = R[c];
  }
}

// ---------------- Kernel A1: update_frames from original inputs ------------
__global__ __launch_bounds__(64) void frames_init_kernel(
    const float* __restrict__ t_nodes, const float* __restrict__ O_nodes,
    const float* __restrict__ t_pair, const float* __restrict__ O_pair,
    const float* __restrict__ w, const int* __restrict__ topo,
    float* __restrict__ t_out, float* __restrict__ O_out) {
  __shared__ __align__(16) float sO[KP * 9];
  __shared__ __align__(16) float st[KP * 3];
  __shared__ __align__(16) float sw_[KP];
  __shared__ float red[2 * 13];

  const int row = blockIdx.x;
  const int tid = threadIdx.x;
  const size_t prow = (size_t)row * KP;

  stage_tile<144>(O_pair + prow * 9, sO, tid);
  stage_tile<48>(t_pair + prow * 3, st, tid);
  stage_tile<16>(w + prow, sw_, tid);
#if USE_ASYNC_LDS
  async_wait_all();
#endif
  __syncthreads();

  const int bbase = (row / NREF) * NREF;
  const int jrow = bbase + topo[prow + tid];

  float Op[9], tp[3];
  ld9(sO + tid * 9, Op);
  ld3(st + tid * 3, tp);
  const float wk = sw_[tid];

  float Oj[9], tj[3];
  ld9(O_nodes + (size_t)jrow * 9, Oj);
  ld3(t_nodes + (size_t)jrow * 3, tj);

  // t_ji = -Op^T tp ; tv = tj + Oj t_ji ; Om = Oj @ Op^T
  float tji[3], tmp3[3];
  m_Tvec(Op, tp, tji);
  m_vec(Oj, tji, tmp3);
  float tv[3] = {tj[0] - tmp3[0], tj[1] - tmp3[1], tj[2] - tmp3[2]};
  float Om[9];
  m_mulBT(Oj, Op, Om);

  reduce_finish(red, tid, wk, tv, Om, t_out + (size_t)row * 3, O_out + (size_t)row * 9);
}

// ------- Kernel F: update_pairwise(iter i) fused with update_frames(i+1) ----
__global__ __launch_bounds__(64) void fused_step_kernel(
    const float* __restrict__ t_cur, const float* __restrict__ O_cur,
    const float* __restrict__ t_new, const float* __restrict__ O_new,
    const float* __restrict__ t_pair_in, const float* __restrict__ O_pair_in,
    const float* __restrict__ w, const int* __restrict__ topo,
    float* __restrict__ t_pair_out, float* __restrict__ O_pair_out,
    float* __restrict__ t_node_out, float* __restrict__ O_node_out, int store_pairs) {
  __shared__ __align__(16) float sO[KP * 9];
  __shared__ __align__(16) float st[KP * 3];
  __shared__ __align__(16) float sw_[KP];
  __shared__ float red[2 * 13];

  const int row = blockIdx.x;
  const int tid = threadIdx.x;
  const size_t prow = (size_t)row * KP;

  stage_tile<144>(O_pair_in + prow * 9, sO, tid);
  stage_tile<48>(t_pair_in + prow * 3, st, tid);
  stage_tile<16>(w + prow, sw_, tid);
#if USE_ASYNC_LDS
  async_wait_all();
#endif
  __syncthreads();

  const int bbase = (row / NREF) * NREF;
  const int jrow = bbase + topo[prow + tid];

  float Op[9], tp[3];
  ld9(sO + tid * 9, Op);
  ld3(st + tid * 3, tp);
  const float wk = sw_[tid];

  float Ocj[9], tcj[3], Onj[9], tnj[3];
  ld9(O_cur + (size_t)jrow * 9, Ocj);
  ld3(t_cur + (size_t)jrow * 3, tcj);
  ld9(O_new + (size_t)jrow * 9, Onj);
  ld3(t_new + (size_t)jrow * 3, tnj);

  float Ocn[9], tcn[3], Onn[9], tnn[3];  // uniform across block -> scalarized
  ld9(O_cur + (size_t)row * 9, Ocn);
  ld3(t_cur + (size_t)row * 3, tcn);
  ld9(O_new + (size_t)row * 9, Onn);
  ld3(t_new + (size_t)row * 3, tnn);

  // ---- update_pairwise ----
  float Ojj[9];
  m_ATmul(Ocj, Onj, Ojj);                       // Ocj^T @ Onj
  float dtj[3] = {tnj[0] - tcj[0], tnj[1] - tcj[1], tnj[2] - tcj[2]};
  float tjj[3];
  m_Tvec(Ocj, dtj, tjj);                        // Ocj^T (tnj - tcj)

  float Oij2[9], tmp3[3];
  m_mul(Op, Ojj, Oij2);
  m_vec(Op, tjj, tmp3);
  float tij2[3] = {tp[0] + tmp3[0], tp[1] + tmp3[1], tp[2] + tmp3[2]};

  float Oii[9];
  m_ATmul(Onn, Ocn, Oii);                       // Onn^T @ Ocn
  float dtn[3] = {tcn[0] - tnn[0], tcn[1] - tnn[1], tcn[2] - tnn[2]};
  float tii[3];
  m_Tvec(Onn, dtn, tii);                        // Onn^T (tcn - tnn)

  float Opn[9];
  m_mul(Oii, Oij2, Opn);
  m_vec(Oii, tij2, tmp3);
  float tpn[3] = {tii[0] + tmp3[0], tii[1] + tmp3[1], tii[2] + tmp3[2]};

  if (store_pairs) {                            // streaming output: non-temporal
    float* tpo = t_pair_out + (prow + tid) * 3;
    float* Opo = O_pair_out + (prow + tid) * 9;
#pragma unroll
    for (int c = 0; c < 3; ++c) __builtin_nontemporal_store(tpn[c], tpo + c);
#pragma unroll
    for (int c = 0; c < 9; ++c) __builtin_nontemporal_store(Opn[c], Opo + c);
  }

  // ---- update_frames for next iteration (new node frames + updated pairs) ----
  float tji[3];
  m_Tvec(Opn, tpn, tji);                        // Opn^T tpn
  m_vec(Onj, tji, tmp3);                        // Onj @ (Opn^T tpn)
  float tv[3] = {tnj[0] - tmp3[0], tnj[1] - tmp3[1], tnj[2] - tmp3[2]};
  float Om[9];
  m_mulBT(Onj, Opn, Om);                        // Onj @ Opn^T

  reduce_finish(red, tid, wk, tv, Om, t_node_out + (size_t)row * 3,
                O_node_out + (size_t)row * 9);
}

// ---------------------------------------------------------------------------
extern "C" void kernel_launch(void* const* d_in, const int* in_sizes, int n_in,
                              void* d_out, int out_size, void* d_ws, size_t ws_size,
                              hipStream_t stream) {
  (void)n_in; (void)out_size; (void)ws_size;
  const float* t_in  = (const float*)d_in[0];  // [B,N,3]
  const float* O_in  = (const float*)d_in[1];  // [B,N,3,3]
  const float* tp_in = (const float*)d_in[2];  // [B,N,K,3]
  const float* Op_in = (const float*)d_in[3];  // [B,N,K,3,3]
  const float* w     = (const float*)d_in[4];  // [B,N,K]
  const int*   topo  = (const int*)d_in[5];    // [B,N,K]

  const int BN = in_sizes[0] / 3;              // B*N = 65536

  float* ws = (float*)d_ws;
  float* tA = ws; ws += (size_t)BN * 3;        // iter-1 node frames
  float* OA = ws; ws += (size_t)BN * 9;
  float* tB = ws; ws += (size_t)BN * 3;        // iter-2 node frames
  float* OB = ws; ws += (size_t)BN * 9;
  float* tP = ws; ws += (size_t)BN * KP * 3;   // live pair state (after iter-1 update)
  float* OP = ws;                              // BN*KP*9

  float* t_out = (float*)d_out;                // output t then O, concatenated
  float* O_out = t_out + (size_t)BN * 3;

  dim3 grid(BN), block(64);
  // iter 1: A1
  frames_init_kernel<<<grid, block, 0, stream>>>(t_in, O_in, tp_in, Op_in, w, topo, tA, OA);
  // iter 1 pairwise + iter 2 frames (writes pair state once)
  fused_step_kernel<<<grid, block, 0, stream>>>(t_in, O_in, tA, OA, tp_in, Op_in, w, topo,
                                                tP, OP, tB, OB, 1);
  // iter 2 pairwise + iter 3 frames (pair output dead -> skipped), write d_out
  fused_step_kernel<<<grid, block, 0, stream>>>(tA, OA, tB, OB, tP, OP, w, topo,
                                                nullptr, nullptr, t_out, O_out, 0);
}